// Circuit_26654567039463
// MI455X (gfx1250) — compile-verified
//
#include <hip/hip_runtime.h>

#define N_MODES   64
#define IN_MODES  48
#define BATCH     2048
#define EVALPTS   100
#define NSUB      5
#define MTILE     16
#define LSTRIDE   68                 // padded row stride (floats) -> conflict-free WMMA A loads
#define PLANE     (MTILE * LSTRIDE)  // 1088 floats per real/imag plane
#define BUFSZ     (2 * PLANE)        // one complex buffer (planar re/im)

typedef __attribute__((ext_vector_type(2))) float v2f;
typedef __attribute__((ext_vector_type(8))) float v8f;

__device__ __forceinline__ float2 cxmul(float2 a, float2 b) {
    return make_float2(a.x * b.x - a.y * b.y, a.x * b.y + a.y * b.x);
}

// ---------------------------------------------------------------------------
// Kernel 1: build W = T2^T (64x64 complex) once.  Single workgroup, all in LDS.
//   U   = (I + iH)^-1 (I - iH)          (Cayley)
//   UtU = U^T U   (plain transpose)
//   mix = UtU (I - UtU + 1e-8 I)^-1
//   T2  = -kappa[col] * (0.5 I + mix);  W[j][i] = T2[i][j]
// Output: wsW[0..4095] = Re(W) row-major [k][n], wsW[4096..8191] = Im(W).
// ---------------------------------------------------------------------------
__device__ void gauss_jordan(float2* M, float2* V, int tid) {
    __shared__ int piv;
    __shared__ float2 fac[64];
    for (int p = 0; p < 64; ++p) {
        if (tid == 0) {
            int best = p;
            float2 e0 = M[p * 64 + p];
            float bv = e0.x * e0.x + e0.y * e0.y;
            for (int r = p + 1; r < 64; ++r) {
                float2 e = M[r * 64 + p];
                float v = e.x * e.x + e.y * e.y;
                if (v > bv) { bv = v; best = r; }
            }
            piv = best;
        }
        __syncthreads();
        int pr = piv;
        if (pr != p) {
            for (int c = tid; c < 64; c += 256) {
                float2 t = M[p * 64 + c]; M[p * 64 + c] = M[pr * 64 + c]; M[pr * 64 + c] = t;
                t = V[p * 64 + c]; V[p * 64 + c] = V[pr * 64 + c]; V[pr * 64 + c] = t;
            }
        }
        __syncthreads();
        float2 d = M[p * 64 + p];
        __syncthreads();  // everyone reads pivot before it is scaled
        float den = d.x * d.x + d.y * d.y + 1e-30f;
        float2 inv = make_float2(d.x / den, -d.y / den);
        for (int c = tid; c < 64; c += 256) {
            M[p * 64 + c] = cxmul(M[p * 64 + c], inv);
            V[p * 64 + c] = cxmul(V[p * 64 + c], inv);
        }
        __syncthreads();
        for (int r = tid; r < 64; r += 256) fac[r] = M[r * 64 + p];
        __syncthreads();
        for (int idx = tid; idx < 4096; idx += 256) {
            int r = idx >> 6, c = idx & 63;
            if (r != p) {
                float2 f = fac[r];
                float2 mp = M[p * 64 + c];
                float2 vp = V[p * 64 + c];
                M[idx] = make_float2(M[idx].x - (f.x * mp.x - f.y * mp.y),
                                     M[idx].y - (f.x * mp.y + f.y * mp.x));
                V[idx] = make_float2(V[idx].x - (f.x * vp.x - f.y * vp.y),
                                     V[idx].y - (f.x * vp.y + f.y * vp.x));
            }
        }
        __syncthreads();
    }
}

__global__ __launch_bounds__(256) void build_w_kernel(const float* __restrict__ params,
                                                      const float* __restrict__ kappa,
                                                      float* __restrict__ wsW) {
    extern __shared__ float sm[];
    float*  Hs = sm;                       // 4096 f
    float2* cA = (float2*)(sm + 4096);     // 4096 c
    float2* cB = cA + 4096;
    float2* cU = cB + 4096;
    float2* cT = cU + 4096;
    const int tid = threadIdx.x;

    // H = 0.5 (M + M^T), M = reshape(concat(params, 0))
    for (int idx = tid; idx < 4096; idx += 256) {
        int r = idx >> 6, c = idx & 63;
        float a = (idx < 4095) ? params[idx] : 0.f;
        int idt = c * 64 + r;
        float b = (idt < 4095) ? params[idt] : 0.f;
        Hs[idx] = 0.5f * (a + b);
    }
    __syncthreads();
    for (int idx = tid; idx < 4096; idx += 256) {
        int r = idx >> 6, c = idx & 63;
        cA[idx] = make_float2((r == c) ? 1.f : 0.f, Hs[idx]);     // I + iH
        cB[idx] = make_float2((r == c) ? 1.f : 0.f, 0.f);
    }
    __syncthreads();
    gauss_jordan(cA, cB, tid);             // cB = (I+iH)^-1
    // U = cB (I - iH)
    for (int idx = tid; idx < 4096; idx += 256) {
        int r = idx >> 6, c = idx & 63;
        float accr = cB[idx].x, acci = cB[idx].y;
        for (int k = 0; k < 64; ++k) {
            float h = Hs[k * 64 + c];
            float2 v = cB[r * 64 + k];      // v * (-i h) = h*v.y - i h*v.x
            accr += v.y * h;
            acci -= v.x * h;
        }
        cU[idx] = make_float2(accr, acci);
    }
    __syncthreads();
    // UtU[i][j] = sum_k U[k][i] U[k][j]   (plain transpose)
    for (int idx = tid; idx < 4096; idx += 256) {
        int i = idx >> 6, j = idx & 63;
        float accr = 0.f, acci = 0.f;
        for (int k = 0; k < 64; ++k) {
            float2 a = cU[k * 64 + i], b = cU[k * 64 + j];
            accr += a.x * b.x - a.y * b.y;
            acci += a.x * b.y + a.y * b.x;
        }
        cT[idx] = make_float2(accr, acci);
    }
    __syncthreads();
    for (int idx = tid; idx < 4096; idx += 256) {
        int r = idx >> 6, c = idx & 63;
        float dr = (r == c) ? (1.f + 1e-8f) : 0.f;
        cA[idx] = make_float2(dr - cT[idx].x, -cT[idx].y);        // I - UtU + eps I
        cB[idx] = make_float2((r == c) ? 1.f : 0.f, 0.f);
    }
    __syncthreads();
    gauss_jordan(cA, cB, tid);             // cB = (I - UtU + eps)^-1
    // mix = UtU @ Ginv  -> cU (U no longer needed)
    for (int idx = tid; idx < 4096; idx += 256) {
        int i = idx >> 6, j = idx & 63;
        float accr = 0.f, acci = 0.f;
        for (int k = 0; k < 64; ++k) {
            float2 a = cT[i * 64 + k], b = cB[k * 64 + j];
            accr += a.x * b.x - a.y * b.y;
            acci += a.x * b.y + a.y * b.x;
        }
        cU[idx] = make_float2(accr, acci);
    }
    __syncthreads();
    // W[j][i] = T2[i][j] = -kappa[j] (0.5 dij + mix[i][j])
    for (int idx = tid; idx < 4096; idx += 256) {
        int i = idx >> 6, j = idx & 63;
        float2 m = cU[idx];
        float mr = m.x + ((i == j) ? 0.5f : 0.f);
        float kj = kappa[j];
        wsW[j * 64 + i]        = -kj * mr;
        wsW[4096 + j * 64 + i] = -kj * m.y;
    }
}

// ---------------------------------------------------------------------------
// Kernel 2: persistent RK45 integration. 128 blocks x 128 threads (4 waves).
// Each block owns 16 batch rows; each wave owns one 16-column N-tile and keeps
// its W fragments (Re/Im) in registers for the whole integration.
// Stage-input combines are fused into the WMMA epilogue (each (m,n) element is
// exclusively owned by one thread in the C-layout) -> 6 barriers per substep.
// ---------------------------------------------------------------------------
__global__ __launch_bounds__(128) void circuit_kernel(const float* __restrict__ A0r,
                                                      const float* __restrict__ A0i,
                                                      const float* __restrict__ biasr,
                                                      const float* __restrict__ biasi,
                                                      const float* __restrict__ omega,
                                                      const float* __restrict__ nlin,
                                                      const float* __restrict__ wsW,
                                                      float* __restrict__ out) {
    extern __shared__ float sm[];
    float* bufy  = sm;                   // state y
    float* bufYa = sm + BUFSZ;           // ping-pong stage inputs
    float* bufYb = sm + 2 * BUFSZ;
    float* bk1   = sm + 3 * BUFSZ;       // k1..k5 (k6 never stored)
    float* bk2   = sm + 4 * BUFSZ;
    float* bk3   = sm + 5 * BUFSZ;
    float* bk4   = sm + 6 * BUFSZ;
    float* bk5   = sm + 7 * BUFSZ;

    const int tid    = threadIdx.x;
    const int lane   = tid & 31;
    const int wid    = tid >> 5;         // wave id 0..3 -> N-tile
    const int m_lane = lane & 15;
    const int khalf  = lane >> 4;
    const int ncol   = wid * 16 + m_lane;
    const int rowbase = blockIdx.x * MTILE;

    const float om  = omega[ncol];
    const float nlc = nlin[ncol];

    // B fragments: W[k][n], k = 4*kk + 2*khalf + v, n = ncol (resident in VGPRs)
    v2f wr[16], wi[16];
    #pragma unroll
    for (int kk = 0; kk < 16; ++kk) {
        int k0 = 4 * kk + 2 * khalf;
        wr[kk][0] = wsW[(k0 + 0) * 64 + ncol];
        wr[kk][1] = wsW[(k0 + 1) * 64 + ncol];
        wi[kk][0] = wsW[4096 + (k0 + 0) * 64 + ncol];
        wi[kk][1] = wsW[4096 + (k0 + 1) * 64 + ncol];
    }

    float2* out2 = (float2*)out;

    // initial state + t=0 snapshot
    for (int e = tid; e < MTILE * 64; e += 128) {
        int m = e >> 6, c = e & 63;
        int g = rowbase + m;
        float vr, vi;
        if (c < IN_MODES) { vr = A0r[g * IN_MODES + c]; vi = A0i[g * IN_MODES + c]; }
        else              { vr = biasr[c - IN_MODES];   vi = biasi[c - IN_MODES];   }
        int off = m * LSTRIDE + c;
        bufy[off] = vr;
        bufy[PLANE + off] = vi;
        out2[(size_t)g * 64 + c] = make_float2(vr, vi);
    }
    __syncthreads();

    const float dt = 0.5f / (float)(EVALPTS - 1);
    const float h  = dt / (float)NSUB;

    // h-scaled Dormand-Prince coefficients
    const float ha21 = h * 0.2f;
    const float ha31 = h * 0.075f,             ha32 = h * 0.225f;
    const float ha41 = h * (44.f/45.f),        ha42 = h * (-56.f/15.f),     ha43 = h * (32.f/9.f);
    const float ha51 = h * (19372.f/6561.f),   ha52 = h * (-25360.f/2187.f),
                ha53 = h * (64448.f/6561.f),   ha54 = h * (-212.f/729.f);
    const float ha61 = h * (9017.f/3168.f),    ha62 = h * (-355.f/33.f),
                ha63 = h * (46732.f/5247.f),   ha64 = h * (49.f/176.f),     ha65 = h * (-5103.f/18656.f);
    const float hb1  = h * (35.f/384.f),       hb3  = h * (500.f/1113.f),
                hb4  = h * (125.f/192.f),      hb5  = h * (-2187.f/6784.f), hb6  = h * (11.f/84.f);

    // complex GEMM part of f(S): 4 independent WMMA chains of 16
    auto wmma_acc = [&](const float* S, v8f& cr0, v8f& cr1, v8f& ci0, v8f& ci1) {
        cr0 = v8f{}; cr1 = v8f{}; ci0 = v8f{}; ci1 = v8f{};
        const float* Sr = S;
        const float* Si = S + PLANE;
        #pragma unroll
        for (int kk = 0; kk < 16; ++kk) {
            int a_off = m_lane * LSTRIDE + 4 * kk + 2 * khalf;
            v2f ar  = *(const v2f*)(Sr + a_off);
            v2f ai  = *(const v2f*)(Si + a_off);
            v2f nai = -ai;
            cr0 = __builtin_amdgcn_wmma_f32_16x16x4_f32(false, ar,  false, wr[kk], (short)0, cr0, false, false);
            cr1 = __builtin_amdgcn_wmma_f32_16x16x4_f32(false, nai, false, wi[kk], (short)0, cr1, false, false);
            ci0 = __builtin_amdgcn_wmma_f32_16x16x4_f32(false, ar,  false, wi[kk], (short)0, ci0, false, false);
            ci1 = __builtin_amdgcn_wmma_f32_16x16x4_f32(false, ai,  false, wr[kk], (short)0, ci1, false, false);
        }
    };

    v8f cr0, cr1, ci0, ci1;

    for (int t = 0; t < EVALPTS - 1; ++t) {
        for (int s = 0; s < NSUB; ++s) {
            // ---- stage 1: k1 = f(y); Y2 = y + h a21 k1 -> Ya
            wmma_acc(bufy, cr0, cr1, ci0, ci1);
            #pragma unroll
            for (int v = 0; v < 8; ++v) {
                const int off = (v + 8 * khalf) * LSTRIDE + ncol;
                const float yr = bufy[off], yi = bufy[PLANE + off];
                const float w  = om + nlc * (yr * yr + yi * yi);
                const float kr = (cr0[v] + cr1[v]) - w * yi;
                const float ki = (ci0[v] + ci1[v]) + w * yr;
                bk1[off] = kr; bk1[PLANE + off] = ki;
                bufYa[off]         = yr + ha21 * kr;
                bufYa[PLANE + off] = yi + ha21 * ki;
            }
            __syncthreads();
            // ---- stage 2: k2 = f(Ya); Y3 -> Yb
            wmma_acc(bufYa, cr0, cr1, ci0, ci1);
            #pragma unroll
            for (int v = 0; v < 8; ++v) {
                const int off = (v + 8 * khalf) * LSTRIDE + ncol;
                const float srv = bufYa[off], siv = bufYa[PLANE + off];
                const float w  = om + nlc * (srv * srv + siv * siv);
                const float kr = (cr0[v] + cr1[v]) - w * siv;
                const float ki = (ci0[v] + ci1[v]) + w * srv;
                bk2[off] = kr; bk2[PLANE + off] = ki;
                bufYb[off]         = bufy[off]         + ha31 * bk1[off]         + ha32 * kr;
                bufYb[PLANE + off] = bufy[PLANE + off] + ha31 * bk1[PLANE + off] + ha32 * ki;
            }
            __syncthreads();
            // ---- stage 3: k3 = f(Yb); Y4 -> Ya
            wmma_acc(bufYb, cr0, cr1, ci0, ci1);
            #pragma unroll
            for (int v = 0; v < 8; ++v) {
                const int off = (v + 8 * khalf) * LSTRIDE + ncol;
                const float srv = bufYb[off], siv = bufYb[PLANE + off];
                const float w  = om + nlc * (srv * srv + siv * siv);
                const float kr = (cr0[v] + cr1[v]) - w * siv;
                const float ki = (ci0[v] + ci1[v]) + w * srv;
                bk3[off] = kr; bk3[PLANE + off] = ki;
                bufYa[off]         = bufy[off]         + ha41 * bk1[off]         + ha42 * bk2[off]         + ha43 * kr;
                bufYa[PLANE + off] = bufy[PLANE + off] + ha41 * bk1[PLANE + off] + ha42 * bk2[PLANE + off] + ha43 * ki;
            }
            __syncthreads();
            // ---- stage 4: k4 = f(Ya); Y5 -> Yb
            wmma_acc(bufYa, cr0, cr1, ci0, ci1);
            #pragma unroll
            for (int v = 0; v < 8; ++v) {
                const int off = (v + 8 * khalf) * LSTRIDE + ncol;
                const float srv = bufYa[off], siv = bufYa[PLANE + off];
                const float w  = om + nlc * (srv * srv + siv * siv);
                const float kr = (cr0[v] + cr1[v]) - w * siv;
                const float ki = (ci0[v] + ci1[v]) + w * srv;
                bk4[off] = kr; bk4[PLANE + off] = ki;
                bufYb[off]         = bufy[off]         + ha51 * bk1[off]         + ha52 * bk2[off]
                                                       + ha53 * bk3[off]         + ha54 * kr;
                bufYb[PLANE + off] = bufy[PLANE + off] + ha51 * bk1[PLANE + off] + ha52 * bk2[PLANE + off]
                                                       + ha53 * bk3[PLANE + off] + ha54 * ki;
            }
            __syncthreads();
            // ---- stage 5: k5 = f(Yb); Y6 -> Ya
            wmma_acc(bufYb, cr0, cr1, ci0, ci1);
            #pragma unroll
            for (int v = 0; v < 8; ++v) {
                const int off = (v + 8 * khalf) * LSTRIDE + ncol;
                const float srv = bufYb[off], siv = bufYb[PLANE + off];
                const float w  = om + nlc * (srv * srv + siv * siv);
                const float kr = (cr0[v] + cr1[v]) - w * siv;
                const float ki = (ci0[v] + ci1[v]) + w * srv;
                bk5[off] = kr; bk5[PLANE + off] = ki;
                bufYa[off]         = bufy[off]         + ha61 * bk1[off]         + ha62 * bk2[off]
                                   + ha63 * bk3[off]   + ha64 * bk4[off]         + ha65 * kr;
                bufYa[PLANE + off] = bufy[PLANE + off] + ha61 * bk1[PLANE + off] + ha62 * bk2[PLANE + off]
                                   + ha63 * bk3[PLANE + off] + ha64 * bk4[PLANE + off] + ha65 * ki;
            }
            __syncthreads();
            // ---- stage 6: k6 = f(Ya); y <- y + h(b1 k1 + b3 k3 + b4 k4 + b5 k5 + b6 k6)
            wmma_acc(bufYa, cr0, cr1, ci0, ci1);
            #pragma unroll
            for (int v = 0; v < 8; ++v) {
                const int off = (v + 8 * khalf) * LSTRIDE + ncol;
                const float srv = bufYa[off], siv = bufYa[PLANE + off];
                const float w  = om + nlc * (srv * srv + siv * siv);
                const float kr = (cr0[v] + cr1[v]) - w * siv;
                const float ki = (ci0[v] + ci1[v]) + w * srv;
                bufy[off]         += hb1 * bk1[off]         + hb3 * bk3[off]
                                   + hb4 * bk4[off]         + hb5 * bk5[off]         + hb6 * kr;
                bufy[PLANE + off] += hb1 * bk1[PLANE + off] + hb3 * bk3[PLANE + off]
                                   + hb4 * bk4[PLANE + off] + hb5 * bk5[PLANE + off] + hb6 * ki;
            }
            __syncthreads();
        }
        for (int e = tid; e < MTILE * 64; e += 128) {             // trajectory snapshot
            int m = e >> 6, c = e & 63;
            int off = m * LSTRIDE + c;
            int g = rowbase + m;
            out2[((size_t)(t + 1) * BATCH + g) * 64 + c] =
                make_float2(bufy[off], bufy[PLANE + off]);
        }
    }
}

// ---------------------------------------------------------------------------
extern "C" void kernel_launch(void* const* d_in, const int* in_sizes, int n_in,
                              void* d_out, int out_size, void* d_ws, size_t ws_size,
                              hipStream_t stream) {
    const float* A0r    = (const float*)d_in[0];
    const float* A0i    = (const float*)d_in[1];
    const float* params = (const float*)d_in[2];
    const float* biasr  = (const float*)d_in[3];
    const float* biasi  = (const float*)d_in[4];
    const float* omega  = (const float*)d_in[5];
    const float* kappa  = (const float*)d_in[6];
    const float* nlin   = (const float*)d_in[7];
    float* out = (float*)d_out;
    float* wsW = (float*)d_ws;     // Wr[4096] ++ Wi[4096]

    const int lds1 = (4096 + 4 * 4096 * 2) * (int)sizeof(float);   // 147456 B
    const int lds2 = 8 * BUFSZ * (int)sizeof(float);               // 69632 B
    (void)hipFuncSetAttribute((const void*)build_w_kernel,
                              hipFuncAttributeMaxDynamicSharedMemorySize, lds1);
    (void)hipFuncSetAttribute((const void*)circuit_kernel,
                              hipFuncAttributeMaxDynamicSharedMemorySize, lds2);

    hipLaunchKernelGGL(build_w_kernel, dim3(1), dim3(256), lds1, stream,
                       params, kappa, wsW);
    hipLaunchKernelGGL(circuit_kernel, dim3(BATCH / MTILE), dim3(128), lds2, stream,
                       A0r, A0i, biasr, biasi, omega, nlin, wsW, out);
}